// SetCriterion_87308095193844
// MI455X (gfx1250) — compile-verified
//
#include <hip/hip_runtime.h>
#include <math.h>

#define BATCH 16
#define NQ    1024
#define NG    64
#define NCLS  92              // C+1
#define ROWS  (BATCH * NQ)    // 16384
#define NBLK  64
#define EPSV  1e-7f

typedef __attribute__((ext_vector_type(2))) float v2f;
typedef __attribute__((ext_vector_type(8))) float v8f;

// ---------------------------------------------------------------------------
// Kernel 1: weighted cross-entropy partials.
// One wave32 per row of 92 logits: lane L holds logits {L, L+32, L+64(<92)}.
// Butterfly shuffles for row max / exp-sum; label logit via uniform __shfl.
// Streams pred_class exactly once (memory-bound: ~6 MB).
// ---------------------------------------------------------------------------
__global__ __launch_bounds__(256) void class_loss_partial(
    const float* __restrict__ pred_class,
    const int*   __restrict__ row_inds,
    const int*   __restrict__ gt_class,
    const float* __restrict__ empty_weight,
    float* __restrict__ ws_num, float* __restrict__ ws_den)
{
    __shared__ float s_num[8], s_den[8];
    const int lane = threadIdx.x & 31;
    const int wid  = threadIdx.x >> 5;
    const int gw   = blockIdx.x * 8 + wid;
    const int nw   = gridDim.x * 8;

    float num = 0.0f, den = 0.0f;
    for (int r = gw; r < ROWS; r += nw) {
        const float* rowp = pred_class + (long)r * NCLS;
        if (r + nw < ROWS)  // speculative streaming prefetch of next row
            __builtin_prefetch(pred_class + (long)(r + nw) * NCLS + lane, 0, 0);

        float x0 = rowp[lane];
        float x1 = rowp[lane + 32];
        float x2 = (lane < 28) ? rowp[lane + 64] : -INFINITY;

        float m = fmaxf(fmaxf(x0, x1), x2);
        for (int off = 16; off > 0; off >>= 1)
            m = fmaxf(m, __shfl_xor(m, off, 32));

        float s = __expf(x0 - m) + __expf(x1 - m) +
                  ((lane < 28) ? __expf(x2 - m) : 0.0f);
        for (int off = 16; off > 0; off >>= 1)
            s += __shfl_xor(s, off, 32);

        float lse = m + __logf(s);

        const int b     = r >> 10;               // r / NQ
        const int rm    = row_inds[r];
        const int label = (rm >= 0) ? gt_class[b * NG + rm] : 0;

        const int   slot = label >> 5;           // wave-uniform
        const float v    = (slot == 0) ? x0 : ((slot == 1) ? x1 : x2);
        const float xl   = __shfl(v, label & 31, 32);
        const float w    = empty_weight[label];

        num += w * (lse - xl);
        den += w;
    }

    if (lane == 0) { s_num[wid] = num; s_den[wid] = den; }
    __syncthreads();
    if (threadIdx.x == 0) {
        float n = 0.0f, d = 0.0f;
        for (int i = 0; i < 8; ++i) { n += s_num[i]; d += s_den[i]; }
        ws_num[blockIdx.x] = n;
        ws_den[blockIdx.x] = d;
    }
}

// ---------------------------------------------------------------------------
// Kernel 2: per-query L1 + diagonal GIoU + background count partials.
// Only the diagonal of the [B,Q,Q] GIoU is ever read by the reference, so we
// compute GIoU(pred_m[b,q], gt_m[b,q]) directly. Unmatched queries produce
// all-zero boxes -> 0 contribution (EPS guards), matching the reference.
// ---------------------------------------------------------------------------
__global__ __launch_bounds__(256) void box_loss_partial(
    const float* __restrict__ pred_bbox,
    const float* __restrict__ gt_bbox,
    const int*   __restrict__ row_inds,
    const int*   __restrict__ gt_class,
    float* __restrict__ ws_l1, float* __restrict__ ws_giou, float* __restrict__ ws_bg)
{
    __shared__ float s_l1[8], s_gi[8], s_bg[8];
    const int t = blockIdx.x * 256 + threadIdx.x;   // one thread per query

    float l1 = 0.0f, gi = 0.0f, bg = 0.0f;
    if (t < ROWS) {
        const int  b     = t >> 10;
        const int  rm    = row_inds[t];
        const int  label = (rm >= 0) ? gt_class[b * NG + rm] : 0;
        const bool match = (rm < 0);
        const bool obj   = (label == 0);
        bg = obj ? 1.0f : 0.0f;

        float p0 = 0.f, p1 = 0.f, p2 = 0.f, p3 = 0.f;
        float g0 = 0.f, g1 = 0.f, g2 = 0.f, g3 = 0.f;
        if (!(match || obj)) {
            const float* pp = pred_bbox + (long)t * 4;
            p0 = pp[0]; p1 = pp[1]; p2 = pp[2]; p3 = pp[3];
        }
        if (!match && rm >= 0) {
            const float* gp = gt_bbox + (long)(b * NG + rm) * 4;
            g0 = gp[0]; g1 = gp[1]; g2 = gp[2]; g3 = gp[3];
        }

        l1 = fabsf(p0 - g0) + fabsf(p1 - g1) + fabsf(p2 - g2) + fabsf(p3 - g3);

        const float area1 = (p2 - p0) * (p3 - p1);
        const float area2 = (g2 - g0) * (g3 - g1);
        const float wx    = fmaxf(fminf(p2, g2) - fmaxf(p0, g0), 0.0f);
        const float wy    = fmaxf(fminf(p3, g3) - fmaxf(p1, g1), 0.0f);
        const float inter = wx * wy;
        const float uni   = area1 + area2 - inter;
        const float iou   = inter / fmaxf(uni, EPSV);
        const float ex    = fmaxf(fmaxf(p2, g2) - fminf(p0, g0), 0.0f);
        const float ey    = fmaxf(fmaxf(p3, g3) - fminf(p1, g1), 0.0f);
        const float enc   = ex * ey;
        gi = iou - (enc - uni) / fmaxf(enc, EPSV);
    }

    for (int off = 16; off > 0; off >>= 1) {
        l1 += __shfl_xor(l1, off, 32);
        gi += __shfl_xor(gi, off, 32);
        bg += __shfl_xor(bg, off, 32);
    }
    const int lane = threadIdx.x & 31, wid = threadIdx.x >> 5;
    if (lane == 0) { s_l1[wid] = l1; s_gi[wid] = gi; s_bg[wid] = bg; }
    __syncthreads();
    if (threadIdx.x == 0) {
        float a = 0.f, g = 0.f, c = 0.f;
        for (int i = 0; i < 8; ++i) { a += s_l1[i]; g += s_gi[i]; c += s_bg[i]; }
        ws_l1[blockIdx.x]   = a;
        ws_giou[blockIdx.x] = g;
        ws_bg[blockIdx.x]   = c;
    }
}

// ---------------------------------------------------------------------------
// Kernel 3: final combine, single wave32 (EXEC all ones, required for WMMA).
// Each 64-float partial array is summed EXACTLY in f32 by one
// V_WMMA_F32_16X16X4_F32: A = 16x4 tile of partials, B = all-ones, so every
// D[m,n] = rowsum(m); lanes 0-15 hold rows 0-7, lanes 16-31 rows 8-15.
// A-layout (ISA 7.12.2, 32-bit 16x4): lane L<16 -> (M=L,K=0),(M=L,K=1);
// lane L>=16 -> (M=L-16,K=2),(M=L-16,K=3). Flat map f(M,K)=K*16+M.
// ---------------------------------------------------------------------------
__global__ __launch_bounds__(32) void final_combine(
    const float* __restrict__ ws, float* __restrict__ out)
{
    const int lane = threadIdx.x;
    const int i0   = (lane < 16) ? lane : (lane + 16);

    v2f bones; bones.x = 1.0f; bones.y = 1.0f;

    float sums[5];
#pragma unroll
    for (int qn = 0; qn < 5; ++qn) {
        const float* vals = ws + qn * NBLK;
        v2f a;
        a.x = vals[i0];
        a.y = vals[i0 + 16];
        v8f c = {};
        c = __builtin_amdgcn_wmma_f32_16x16x4_f32(
                /*neg_a=*/false, a, /*neg_b=*/false, bones,
                /*c_mod=*/(short)0, c, /*reuse_a=*/false, /*reuse_b=*/false);
        float t = c[0] + c[1] + c[2] + c[3] + c[4] + c[5] + c[6] + c[7];
        sums[qn] = t + __shfl_xor(t, 16, 32);   // rows 0-7 + rows 8-15
    }

    const float class_loss = sums[0] / sums[1];
    const float bbox_loss  = sums[2] / fmaxf(sums[4], 1.0f);
    const float giou_loss  = 1.0f - sums[3] / (float)ROWS;
    const float total = class_loss + 5.0f * bbox_loss + 2.0f * giou_loss;

    if (lane == 0) out[0] = total;
}

// ---------------------------------------------------------------------------
extern "C" void kernel_launch(void* const* d_in, const int* in_sizes, int n_in,
                              void* d_out, int out_size, void* d_ws, size_t ws_size,
                              hipStream_t stream)
{
    const float* pred_class   = (const float*)d_in[0];
    const float* pred_bbox    = (const float*)d_in[1];
    const int*   gt_class     = (const int*)d_in[2];
    const float* gt_bbox      = (const float*)d_in[3];
    const int*   row_inds     = (const int*)d_in[4];
    /* col_inds (d_in[5]) unused: row_inds is its exact inverse */
    const float* empty_weight = (const float*)d_in[6];

    float* ws  = (float*)d_ws;   // layout: [num|den|l1|giou|bg] x NBLK floats
    float* out = (float*)d_out;

    class_loss_partial<<<NBLK, 256, 0, stream>>>(
        pred_class, row_inds, gt_class, empty_weight, ws + 0 * NBLK, ws + 1 * NBLK);
    box_loss_partial<<<NBLK, 256, 0, stream>>>(
        pred_bbox, gt_bbox, row_inds, gt_class,
        ws + 2 * NBLK, ws + 3 * NBLK, ws + 4 * NBLK);
    final_combine<<<1, 32, 0, stream>>>(ws, out);
}